// Net_61744449847960
// MI455X (gfx1250) — compile-verified
//
#include <hip/hip_runtime.h>

// RLeaky SNN forward: mem = 0.95*mem + x_t + spk@W^T + b - H(mem-1); spk = H(mem-1)
// B=256, T=128, F=2048.  Outputs: spk_rec [B,T,F] then mem_rec [B,T,F] (fp32).
//
// 128 sequential step-kernels (graph nodes). Each step: fused NT-GEMM
// (R[m,n] = sum_k spk[m,k] * W[n,k]) via exact-fp32 V_WMMA_F32_16X16X4_F32
// + elementwise membrane update. The spike/membrane records in d_out double
// as recurrent state: step t reads step t-1's rows (t==0 -> zero state).
//
// Tiling: block = 128 threads = 4 waves; each wave computes a 16(M) x 32(N)
// strip (two v8f accumulators sharing one A fragment). Block tile 16M x 128N,
// grid = (2048/128, 256/16) = (16,16) = 256 blocks -> spreads 1024 waves at
// 1 wave/SIMD across 256 WGPs. All 4 waves in a block read identical A
// (spike) fragments -> L0 hits; W fragments are unique per wave.

typedef __attribute__((ext_vector_type(2))) float v2f;
typedef __attribute__((ext_vector_type(8))) float v8f;

#define B_  256
#define T_  128
#define F_  2048
#define TF_ (T_ * F_)

#if defined(__has_builtin)
#  if __has_builtin(__builtin_amdgcn_wmma_f32_16x16x4_f32)
#    define HAVE_WMMA_F32 1
#  endif
#endif
#ifndef HAVE_WMMA_F32
#  define HAVE_WMMA_F32 0
#endif

__global__ __launch_bounds__(128) void rleaky_step_kernel(
    const float* __restrict__ x,      // [B,T,F]
    const float* __restrict__ W,      // [F,F] row-major; R[m,n]=sum_k S[m,k]*W[n,k]
    const float* __restrict__ bias,   // [F]
    float* __restrict__ spk_out,      // [B,T,F] (also recurrent spike state)
    float* __restrict__ mem_out,      // [B,T,F] (also recurrent membrane state)
    int t)
{
    const int lane = threadIdx.x & 31;   // wave32
    const int wn   = threadIdx.x >> 5;   // 0..3: which 32-wide N strip
    const int half = lane >> 4;          // 0: lanes 0-15, 1: lanes 16-31
    const int l16  = lane & 15;

    const int mBase = blockIdx.y * 16;               // batch rows (shared by all waves)
    const int nBase = blockIdx.x * 128 + wn * 32;    // feature cols (two 16-wide tiles)

    v8f acc0 = {};   // C tile for N = nBase .. nBase+15
    v8f acc1 = {};   // C tile for N = nBase+16 .. nBase+31

#if HAVE_WMMA_F32
    if (t > 0) {
        // S[m][k] = spk_out[m*TF + (t-1)*F + k]
        const float* __restrict__ arow  =
            spk_out + (size_t)(mBase + l16) * TF_ + (size_t)(t - 1) * F_ + 2 * half;
        const float* __restrict__ brow0 =
            W + (size_t)(nBase + l16) * F_ + 2 * half;
        const float* __restrict__ brow1 =
            W + (size_t)(nBase + 16 + l16) * F_ + 2 * half;

        // K advances by 4 per WMMA (16x16x4 fp32). Fragment layout per ISA:
        //   A vgpr pair = S[mBase+l16][k + 2*half + {0,1}]
        //   B vgpr pair = W[nBase+l16][k + 2*half + {0,1}]
        #pragma unroll 4
        for (int k = 0; k < F_; k += 4) {
            v2f a  = *(const v2f*)(arow + k);
            v2f b0 = *(const v2f*)(brow0 + k);
            v2f b1 = *(const v2f*)(brow1 + k);
            acc0 = __builtin_amdgcn_wmma_f32_16x16x4_f32(
                false, a, false, b0, (short)0, acc0, false, false);
            acc1 = __builtin_amdgcn_wmma_f32_16x16x4_f32(
                false, a, false, b1, (short)0, acc1, false, false);
        }
    }
#endif

    // Epilogue: C layout -> VGPR i holds M = i + 8*half, N = l16 (per tile).
    const size_t tOff  = (size_t)t * F_;
    const size_t tPrev = (size_t)(t - 1) * F_;

    #pragma unroll
    for (int i = 0; i < 8; ++i) {
        const int gm = mBase + i + 8 * half;
        const size_t rowBase = (size_t)gm * TF_;

        #pragma unroll
        for (int tile = 0; tile < 2; ++tile) {
            const int gn = nBase + tile * 16 + l16;

            float r;
#if HAVE_WMMA_F32
            r = tile ? acc1[i] : acc0[i];
#else
            // Scalar fallback (no matrix units): exact same math.
            r = 0.0f;
            if (t > 0) {
                const float* Sr = spk_out + rowBase + tPrev;
                const float* Wr = W + (size_t)gn * F_;
                for (int k = 0; k < F_; ++k) r = fmaf(Sr[k], Wr[k], r);
            }
#endif
            const float memPrev = (t > 0) ? mem_out[rowBase + tPrev + gn] : 0.0f;
            const float reset   = (memPrev > 1.0f) ? 1.0f : 0.0f;   // H(mem - 1)
            const size_t idx    = rowBase + tOff + gn;
            const float memNew  = 0.95f * memPrev + x[idx] + r + bias[gn] - reset;
            const float spkNew  = (memNew > 1.0f) ? 1.0f : 0.0f;

            spk_out[idx] = spkNew;
            mem_out[idx] = memNew;
        }
    }
}

extern "C" void kernel_launch(void* const* d_in, const int* in_sizes, int n_in,
                              void* d_out, int out_size, void* d_ws, size_t ws_size,
                              hipStream_t stream) {
    (void)in_sizes; (void)n_in; (void)out_size; (void)d_ws; (void)ws_size;

    const float* x    = (const float*)d_in[0];   // [B,T,F]
    const float* W    = (const float*)d_in[1];   // [F,F]
    const float* bias = (const float*)d_in[2];   // [F]

    float* spk = (float*)d_out;                  // [B,T,F]
    float* mem = spk + (size_t)B_ * T_ * F_;     // [B,T,F]

    dim3 grid(F_ / 128, B_ / 16);                // (16, 16) = 256 blocks
    for (int t = 0; t < T_; ++t) {
        rleaky_step_kernel<<<grid, dim3(128), 0, stream>>>(x, W, bias, spk, mem, t);
    }
}